// ROIAlign_2559800508427
// MI455X (gfx1250) — compile-verified
//
#include <hip/hip_runtime.h>

// ROIAlign (crop_and_resize 14x14 bilinear + 2x2 maxpool -> 7x7) for MI455X.
//
// Strategy: x-interpolation expressed as a banded matmul on the WMMA engine
// (V_WMMA_F32_16X16X4_F32), y-interpolation as VALU FMA on contiguous rows.
// Feature map (134 MB) fits in the 192 MB L2, so gathers are L2-resident;
// contiguous span loads replace scattered 4-tap gathers at similar byte cost.

typedef __attribute__((ext_vector_type(2))) float v2f;
typedef __attribute__((ext_vector_type(8))) float v8f;

#define POOL_ 14
#define H_ 128
#define W_ 128
#define C_ 256

__global__ __launch_bounds__(512)
void roialign_wmma_kernel(const float* __restrict__ rois,
                          const float* __restrict__ fmap,
                          float* __restrict__ out,
                          int nroi)
{
    const int n    = blockIdx.x;      // roi
    const int oy   = blockIdx.y;      // pooled output row 0..6
    const int lane = threadIdx.x & 31;
    const int wave = threadIdx.x >> 5;
    const int c0   = wave * 16;       // 16-channel slice per wave (16 waves -> 256)
    const int h    = lane >> 4;       // half-wave (selects K pair in 32b fragments)
    const int nl   = lane & 15;       // A-row (output ix) / B,C,D column (channel)

    if (n >= nroi) return;

    // ROI scalars (wave-uniform)
    const float y1 = rois[n*5 + 1];
    const float x1 = rois[n*5 + 2];
    const float y2 = rois[n*5 + 3];
    const float x2 = rois[n*5 + 4];
    const int   b  = (int)rois[n*5 + 0];

    const float sy  = (y2 - y1) * (float)(H_ - 1) / (float)(POOL_ - 1);
    const float sx  = (x2 - x1) * (float)(W_ - 1) / (float)(POOL_ - 1);
    const float oyf = y1 * (float)(H_ - 1);
    const float oxf = x1 * (float)(W_ - 1);

    // x-span of the 14 sample columns (in_x monotone increasing since x2 > x1)
    int xlo = (int)floorf(oxf);
    int xhi = (int)floorf(oxf + 13.0f * sx) + 1;
    xlo = min(max(xlo, 0), W_ - 1);
    xhi = min(max(xhi, 0), W_ - 1);
    const int ntiles = (xhi - xlo + 4) >> 2;   // ceil(span/4), K-tiles of 4

    // Per-lane x-interp weights for A-matrix row nl (rows 14,15 forced to 0).
    const float fx  = oxf + (float)nl * sx;
    const float x0f = floorf(fx);
    const float lx  = fx - x0f;
    const float xv  = (fx >= 0.0f && fx <= (float)(W_ - 1) && nl < POOL_) ? 1.0f : 0.0f;
    const int   xc0 = min(max((int)x0f,     0), W_ - 1);
    const int   xc1 = min(max((int)x0f + 1, 0), W_ - 1);
    const float w0  = (1.0f - lx) * xv;   // weight at column xc0
    const float w1  = lx * xv;            // weight at column xc1 (adds if clamped equal)

    const float* fbase = fmap + (size_t)b * H_ * W_ * C_;

    v8f acc[2];
    #pragma unroll
    for (int iy2 = 0; iy2 < 2; ++iy2) {
        const int   iy  = 2 * oy + iy2;
        const float fy  = oyf + (float)iy * sy;
        const float y0f = floorf(fy);
        const float ly  = fy - y0f;
        const float yv  = (fy >= 0.0f && fy <= (float)(H_ - 1)) ? 1.0f : 0.0f;
        const int   yr0 = min(max((int)y0f,     0), H_ - 1);
        const int   yr1 = min(max((int)y0f + 1, 0), H_ - 1);
        const float wy0 = (1.0f - ly) * yv;
        const float wy1 = ly * yv;

        const float* row0 = fbase + (size_t)yr0 * W_ * C_;
        const float* row1 = fbase + (size_t)yr1 * W_ * C_;

        v8f c = {};
        for (int t = 0; t < ntiles; ++t) {
            // 32-bit fragment K layout: v0 holds K = 2h, v1 holds K = 2h+1.
            const int xa  = xlo + 4 * t + 2 * h;
            const int xb  = xa + 1;
            const int xac = min(xa, W_ - 1);
            const int xbc = min(xb, W_ - 1);

            // B fragment: y-interpolated feature at (column, channel c0+nl).
            const size_t ca = (size_t)xac * C_ + (size_t)(c0 + nl);
            const size_t cb = (size_t)xbc * C_ + (size_t)(c0 + nl);
            v2f bfrag;
            bfrag.x = row0[ca] * wy0 + row1[ca] * wy1;
            bfrag.y = row0[cb] * wy0 + row1[cb] * wy1;

            // A fragment: banded x-interp weights Wx[nl, xa], Wx[nl, xb].
            v2f afrag;
            afrag.x = ((xa == xc0) ? w0 : 0.0f) + ((xa == xc1) ? w1 : 0.0f);
            afrag.y = ((xb == xc0) ? w0 : 0.0f) + ((xb == xc1) ? w1 : 0.0f);

            // D = A(16x4) x B(4x16) + C ; accumulates out14[ix, c] over K tiles.
            c = __builtin_amdgcn_wmma_f32_16x16x4_f32(
                    /*neg_a=*/false, afrag, /*neg_b=*/false, bfrag,
                    /*c_mod=*/(short)0, c, /*reuse_a=*/false, /*reuse_b=*/false);
        }
        acc[iy2] = c;
    }

    // 2x2 max pool directly on the C/D register layout:
    // reg r, half h -> M = r + 8h; pairs (2ox, 2ox+1) are adjacent regs.
    float* obase = out + ((size_t)(n * 7 + oy) * 7) * C_ + (size_t)(c0 + nl);
    #pragma unroll
    for (int r = 0; r < 8; r += 2) {
        const int m  = r + 8 * h;     // == 2*ox
        const int ox = m >> 1;
        const float v = fmaxf(fmaxf(acc[0][r], acc[0][r + 1]),
                              fmaxf(acc[1][r], acc[1][r + 1]));
        if (ox < 7) obase[(size_t)ox * C_] = v;   // coalesced 64B per half-wave
    }
}

extern "C" void kernel_launch(void* const* d_in, const int* in_sizes, int n_in,
                              void* d_out, int out_size, void* d_ws, size_t ws_size,
                              hipStream_t stream) {
    const float* rois = (const float*)d_in[0];   // [N,5] (b_idx, y1, x1, y2, x2)
    const float* fmap = (const float*)d_in[1];   // [8,128,128,256] NHWC f32
    float* out = (float*)d_out;                  // [N,7,7,256] f32

    const int nroi = in_sizes[0] / 5;
    dim3 grid((unsigned)nroi, 7, 1);
    roialign_wmma_kernel<<<grid, 512, 0, stream>>>(rois, fmap, out, nroi);
}